// YOLOLoss_59270548685395
// MI455X (gfx1250) — compile-verified
//
#include <hip/hip_runtime.h>
#include <hip/hip_bf16.h>

// ---------------- problem constants (from reference) ----------------
#define ELE          30            // 2 boxes*5 + 20 classes
#define TOTAL_CELLS  (16384 * 49)  // 802816 flat (batch,cell) pairs
#define TILE         128           // cells staged per tile
#define THREADS      128           // 4 wave32s per block
#define NTILES       (TOTAL_CELLS / TILE)     // 6272
#define NBLOCKS      3136
#define TILES_PER_BLOCK (NTILES / NBLOCKS)    // 2 (enables double buffering)
#define CHUNKS_PER_ARRAY (TILE * ELE / 4)     // 960 float4 per array per tile
#define CHUNKS_TOTAL     (2 * CHUNKS_PER_ARRAY) // 1920
#define OPS_PER_THREAD   (CHUNKS_TOTAL / THREADS) // 15 async b128 per lane/tile

typedef __attribute__((ext_vector_type(2))) float v2f;
typedef __attribute__((ext_vector_type(8))) float v8f;
typedef int v4i_vs __attribute__((vector_size(16)));   // matches builtin param

#define GLOBAL_AS __attribute__((address_space(1)))
#define LOCAL_AS  __attribute__((address_space(3)))

// ---------------- CDNA5 feature detection (compile-safe) ----------------
#if defined(__HIP_DEVICE_COMPILE__) && defined(__AMDGCN__) && \
    __has_builtin(__builtin_amdgcn_global_load_async_to_lds_b128) && \
    __has_builtin(__builtin_amdgcn_s_wait_asynccnt)
#define USE_ASYNC 1
#else
#define USE_ASYNC 0
#endif

#if defined(__HIP_DEVICE_COMPILE__) && defined(__AMDGCN__) && \
    __has_builtin(__builtin_amdgcn_wmma_f32_16x16x4_f32)
#define USE_WMMA 1
#else
#define USE_WMMA 0
#endif

#if USE_ASYNC
#define ASYNC_WAIT(n) __builtin_amdgcn_s_wait_asynccnt(n)
#else
#define ASYNC_WAIT(n) ((void)0)
#endif

// ---------------- tile staging: coalesced global -> LDS ----------------
__device__ __forceinline__ void issue_tile(const float4* __restrict__ Pv,
                                           const float4* __restrict__ Tv,
                                           float4* dst,   // generic ptr into __shared__
                                           long tile, int tid) {
  const long gbase = tile * (long)CHUNKS_PER_ARRAY;  // float4 index, 16B-aligned
#pragma unroll
  for (int i = 0; i < OPS_PER_THREAD; ++i) {
    const int c = i * THREADS + tid;                 // 0..1919
    const float4* src = (c < CHUNKS_PER_ARRAY)
                            ? (Pv + gbase + c)
                            : (Tv + gbase + (c - CHUNKS_PER_ARRAY));
#if USE_ASYNC
    // CDNA5 async DMA: memory -> LDS, no VGPR round trip, tracked by ASYNCcnt.
    // Builtin wants (v4i AS1* gsrc, v4i AS3* lds, imm offset, cpol).
    __builtin_amdgcn_global_load_async_to_lds_b128(
        (GLOBAL_AS v4i_vs*)(uintptr_t)src,
        (LOCAL_AS v4i_vs*)(v4i_vs*)(dst + c), 0, 0);
#else
    dst[c] = *src;
#endif
  }
}

// ---------------- per-cell YOLO loss (all f32, mirrors reference) ------
__device__ __forceinline__ float cell_loss(const float* p, const float* t) {
  const float conf_flag = t[5];
  const float coord = (conf_flag > 0.0f) ? 1.0f : 0.0f;
  const float noobj = (conf_flag == 0.0f) ? 1.0f : 0.0f;

  float pc[2][4], tc[2][4], pa[2], ta[2], dconf[2];
  float noobj_sum = 0.0f;
#pragma unroll
  for (int j = 0; j < 2; ++j) {
    const float pcx = p[5 * j + 0], pcy = p[5 * j + 1];
    const float pw  = p[5 * j + 2], ph  = p[5 * j + 3];
    const float tcx = t[5 * j + 0], tcy = t[5 * j + 1];
    const float tw  = t[5 * j + 2], th  = t[5 * j + 3];
    pc[j][0] = pcx - pw * 0.5f; pc[j][1] = pcy - ph * 0.5f;
    pc[j][2] = pcx + pw * 0.5f; pc[j][3] = pcy + ph * 0.5f;
    tc[j][0] = tcx - tw * 0.5f; tc[j][1] = tcy - th * 0.5f;
    tc[j][2] = tcx + tw * 0.5f; tc[j][3] = tcy + th * 0.5f;
    pa[j] = (pc[j][2] - pc[j][0]) * (pc[j][3] - pc[j][1]);
    ta[j] = (tc[j][2] - tc[j][0]) * (tc[j][3] - tc[j][1]);
    const float d4 = p[5 * j + 4] - t[5 * j + 4];
    dconf[j] = d4;
    noobj_sum += d4 * d4;
  }

  float iou[2][2];
#pragma unroll
  for (int jp = 0; jp < 2; ++jp) {
#pragma unroll
    for (int jt = 0; jt < 2; ++jt) {
      const float lt0 = fmaxf(pc[jp][0], tc[jt][0]);
      const float lt1 = fmaxf(pc[jp][1], tc[jt][1]);
      const float rb0 = fminf(pc[jp][2], tc[jt][2]);
      const float rb1 = fminf(pc[jp][3], tc[jt][3]);
      const float iw = fmaxf(rb0 - lt0, 0.0f);
      const float ih = fmaxf(rb1 - lt1, 0.0f);
      const float inter = iw * ih;
      iou[jp][jt] = inter / (pa[jp] + ta[jt] - inter + 1e-10f);
    }
  }

  // jnp.argmax over the predictor axis: first index wins ties
  const int b0 = (iou[1][0] > iou[0][0]) ? 1 : 0;
  const int b1 = (iou[1][1] > iou[0][1]) ? 1 : 0;
  const float resp0 = coord * (((b0 == 0) || (b1 == 0)) ? 1.0f : 0.0f);
  const float resp1 = coord * (((b0 == 1) || (b1 == 1)) ? 1.0f : 0.0f);

  const float obj_conf = resp0 * dconf[0] * dconf[0] + resp1 * dconf[1] * dconf[1];

  float coordsum = 0.0f;
#pragma unroll
  for (int j = 0; j < 2; ++j) {
    const float r = (j == 0) ? resp0 : resp1;
    const float d0 = p[5 * j + 0] - t[5 * j + 0];
    const float d1 = p[5 * j + 1] - t[5 * j + 1];
    const float d2 = p[5 * j + 2] - t[5 * j + 2];
    const float d3 = p[5 * j + 3] - t[5 * j + 3];
    coordsum += r * (d0 * d0 + d1 * d1 + d2 * d2 + d3 * d3);
  }

  float cls = 0.0f;
#pragma unroll
  for (int k = 10; k < 30; ++k) {
    const float d = p[k] - t[k];
    cls += d * d;
  }

  return 5.0f * coordsum + obj_conf + 0.5f * noobj * noobj_sum + coord * cls;
}

// ---------------- main kernel: stream tiles, partial sum per block -----
__global__ __launch_bounds__(THREADS) void yolo_loss_main(
    const float* __restrict__ P, const float* __restrict__ T,
    float* __restrict__ partials) {
  __shared__ float4 stage[2][CHUNKS_TOTAL];  // 61440 B, double buffered
  __shared__ float red[THREADS];

  const int tid = threadIdx.x;
  const int bid = blockIdx.x;
  const float4* Pv = (const float4*)P;
  const float4* Tv = (const float4*)T;

  float acc = 0.0f;

  // prologue: DMA tile 0 into buffer 0
  issue_tile(Pv, Tv, &stage[0][0], (long)bid * TILES_PER_BLOCK, tid);

  for (int i = 0; i < TILES_PER_BLOCK; ++i) {
    if (i + 1 < TILES_PER_BLOCK) {
      // prefetch next tile into the other buffer, then wait only for the
      // oldest OPS_PER_THREAD ops (this tile) — async loads complete in order
      issue_tile(Pv, Tv, &stage[(i + 1) & 1][0],
                 (long)bid * TILES_PER_BLOCK + i + 1, tid);
      ASYNC_WAIT(OPS_PER_THREAD);
    } else {
      ASYNC_WAIT(0);
    }
    __syncthreads();  // all waves' DMA for this tile landed in LDS

    const float* sP = (const float*)&stage[i & 1][0];
    const float* sT = sP + TILE * ELE;
    acc += cell_loss(sP + tid * ELE, sT + tid * ELE);

    __syncthreads();  // buffer free before it is re-issued
  }

  // block tree reduction
  red[tid] = acc;
  __syncthreads();
  for (int s = THREADS >> 1; s > 0; s >>= 1) {
    if (tid < s) red[tid] += red[tid + s];
    __syncthreads();
  }
  if (tid == 0) partials[bid] = red[0];
}

// ---------------- final reduction: one wave, WMMA cross-lane sum -------
// Ones-matrix trick with V_WMMA_F32_16X16X4_F32 (exact f32 accumulate):
//   pass 1: D1 = A(p,0) x B(ones)        -> r_m = p_m + p_{m+16} in vgpr m%8 of lane half m/8
//   pass 2: D2 += A(ones) x B(c1 pairs)  -> every slot = sum_m r_m = total
__global__ __launch_bounds__(32) void yolo_loss_reduce(
    const float* __restrict__ partials, float* __restrict__ out) {
  const int lane = threadIdx.x;  // exactly one wave32, EXEC all ones
  float s = 0.0f;
  for (int i = lane; i < NBLOCKS; i += 32)  // 3136/32 = 98 uniform trips
    s += partials[i];

#if USE_WMMA
  const v2f ones = {1.0f, 1.0f};
  v2f a = {s, 0.0f};
  v8f z = {};
  v8f c1 = __builtin_amdgcn_wmma_f32_16x16x4_f32(
      false, a, false, ones, (short)0, z, false, false);
  v8f c2 = {};
#pragma unroll
  for (int j = 0; j < 4; ++j) {
    v2f b = {c1[2 * j], c1[2 * j + 1]};
    c2 = __builtin_amdgcn_wmma_f32_16x16x4_f32(
        false, ones, false, b, (short)0, c2, false, false);
  }
  if (lane == 0) out[0] = c2[0];
#else
  for (int off = 16; off > 0; off >>= 1) s += __shfl_down(s, off, 32);
  if (lane == 0) out[0] = s;
#endif
}

// ---------------- entry point ------------------------------------------
extern "C" void kernel_launch(void* const* d_in, const int* in_sizes, int n_in,
                              void* d_out, int out_size, void* d_ws,
                              size_t ws_size, hipStream_t stream) {
  (void)in_sizes; (void)n_in; (void)out_size; (void)ws_size;
  const float* P = (const float*)d_in[0];
  const float* T = (const float*)d_in[1];
  float* partials = (float*)d_ws;  // NBLOCKS floats (~12.5 KB), fully rewritten

  yolo_loss_main<<<NBLOCKS, THREADS, 0, stream>>>(P, T, partials);
  yolo_loss_reduce<<<1, 32, 0, stream>>>(partials, (float*)d_out);
}